// VMambaDiscriminator_61564061221597
// MI455X (gfx1250) — compile-verified
//
#include <hip/hip_runtime.h>
#include <hip/hip_bf16.h>
#include <math.h>

// ---------------------------------------------------------------------------
// VMamba discriminator forward, fp32, CDNA5 (gfx1250) wave32.
// GEMMs: V_WMMA_F32_16X16X4_F32, multi-tile per wave for operand reuse.
// Scan: LDS staging via GLOBAL_LOAD_ASYNC_TO_LDS_B32 (+ s_wait_asynccnt).
// ---------------------------------------------------------------------------

typedef __attribute__((ext_vector_type(2))) float v2f;
typedef __attribute__((ext_vector_type(8))) float v8f;

#define B_    2
#define C_    256
#define IMG_  128
#define P_    4
#define H_    32
#define W_    32
#define L_    1024
#define D_    512
#define N_    16
#define R_    16
#define K_    4
#define KPATCH 4096   // C_ * P_ * P_

__device__ __forceinline__ v8f wmma_f32(v2f a, v2f b, v8f c) {
  return __builtin_amdgcn_wmma_f32_16x16x4_f32(false, a, false, b, (short)0, c,
                                               false, false);
}

__device__ __forceinline__ float sigmoidf_(float x) {
  return 1.0f / (1.0f + __expf(-x));
}
__device__ __forceinline__ float softplusf_(float x) {
  return (x > 20.0f) ? x : __logf(1.0f + __expf(x));
}

// xs gather: scan direction k from xconv (B, D, H, W)
__device__ __forceinline__ float xs_val(const float* __restrict__ xconv,
                                        int b, int k, int d, int l) {
  int ll = (k & 2) ? (L_ - 1 - l) : l;
  int h, w;
  if (k & 1) { w = ll >> 5; h = ll & 31; }      // transposed (w-major) scan
  else       { h = ll >> 5; w = ll & 31; }
  return xconv[((((size_t)b * D_) + d) << 10) + (h << 5) + w];
}

// ---------------------------------------------------------------------------
// K1: patch embed GEMM  t[2048,256] = X_patches[2048,4096] * Wp^T + bias
// One wave computes a 16M x 64N strip (4 accumulators, shared A fragment).
// ---------------------------------------------------------------------------
__global__ __launch_bounds__(256) void k_patch_embed(
    const float* __restrict__ x, const float* __restrict__ pw,
    const float* __restrict__ pb, float* __restrict__ t) {
  int wave = blockIdx.x * (blockDim.x >> 5) + (threadIdx.x >> 5);
  int lane = threadIdx.x & 31;
  int mt = wave >> 2;                // 0..127
  int ng = wave & 3;                 // 0..3  (64-col group)
  int mrowA = mt * 16 + (lane & 15);
  int b  = mrowA >> 10;
  int hw = mrowA & 1023;
  int h = hw >> 5, w = hw & 31;
  const float* xb = x + (size_t)b * C_ * IMG_ * IMG_;
  int khalf = (lane >> 4) << 1;      // 0 or 2
  int ncol  = ng * 64 + (lane & 15);
  const float* w0 = pw + (size_t)(ncol +  0) * KPATCH;
  const float* w1 = pw + (size_t)(ncol + 16) * KPATCH;
  const float* w2 = pw + (size_t)(ncol + 32) * KPATCH;
  const float* w3 = pw + (size_t)(ncol + 48) * KPATCH;
  v8f a0 = {}, a1 = {}, a2 = {}, a3 = {};
#pragma unroll 4
  for (int k0 = 0; k0 < KPATCH; k0 += 4) {
    int k = k0 + khalf;
    int ci = k >> 4, rem = k & 15, ph = rem >> 2, pv = rem & 3;
    v2f a;
    a.x = xb[((ci * IMG_) + (h * 4 + ph)) * IMG_ + (w * 4 + pv)];
    int k1 = k + 1;
    ci = k1 >> 4; rem = k1 & 15; ph = rem >> 2; pv = rem & 3;
    a.y = xb[((ci * IMG_) + (h * 4 + ph)) * IMG_ + (w * 4 + pv)];
    v2f b0, b1, b2, b3;
    b0.x = w0[k]; b0.y = w0[k1];
    b1.x = w1[k]; b1.y = w1[k1];
    b2.x = w2[k]; b2.y = w2[k1];
    b3.x = w3[k]; b3.y = w3[k1];
    a0 = wmma_f32(a, b0, a0);
    a1 = wmma_f32(a, b1, a1);
    a2 = wmma_f32(a, b2, a2);
    a3 = wmma_f32(a, b3, a3);
  }
  int mbase = mt * 16 + ((lane >> 4) ? 8 : 0);
  int nb = ng * 64 + (lane & 15);
#pragma unroll
  for (int r = 0; r < 8; ++r) {
    size_t row = (size_t)(mbase + r) * C_;
    t[row + nb +  0] = a0[r] + pb[nb +  0];
    t[row + nb + 16] = a1[r] + pb[nb + 16];
    t[row + nb + 32] = a2[r] + pb[nb + 32];
    t[row + nb + 48] = a3[r] + pb[nb + 48];
  }
}

// ---------------------------------------------------------------------------
// K2: layernorm over C=256 (one block per row)
// ---------------------------------------------------------------------------
__global__ __launch_bounds__(256) void k_ln256(
    const float* __restrict__ t, const float* __restrict__ g,
    const float* __restrict__ bb, float* __restrict__ u) {
  __shared__ float red[256];
  int row = blockIdx.x, c = threadIdx.x;
  float v = t[(size_t)row * C_ + c];
  red[c] = v; __syncthreads();
  for (int s = 128; s > 0; s >>= 1) { if (c < s) red[c] += red[c + s]; __syncthreads(); }
  float mu = red[0] * (1.0f / C_);
  __syncthreads();
  float dv = v - mu;
  red[c] = dv * dv; __syncthreads();
  for (int s = 128; s > 0; s >>= 1) { if (c < s) red[c] += red[c + s]; __syncthreads(); }
  float var = red[0] * (1.0f / C_);
  u[(size_t)row * C_ + c] = dv * rsqrtf(var + 1e-5f) * g[c] + bb[c];
}

// ---------------------------------------------------------------------------
// K3: xz GEMM  xz[2048,1024] = u[2048,256] * in_w^T   (16M x 64N per wave)
// ---------------------------------------------------------------------------
__global__ __launch_bounds__(256) void k_xz(
    const float* __restrict__ u, const float* __restrict__ inw,
    float* __restrict__ xz) {
  int wave = blockIdx.x * (blockDim.x >> 5) + (threadIdx.x >> 5);
  int lane = threadIdx.x & 31;
  int mt = wave >> 4;                // 0..127
  int ng = wave & 15;                // 0..15
  int mrowA = mt * 16 + (lane & 15);
  int khalf = (lane >> 4) << 1;
  int ncol  = ng * 64 + (lane & 15);
  const float* urow = u + (size_t)mrowA * C_;
  const float* w0 = inw + (size_t)(ncol +  0) * C_;
  const float* w1 = inw + (size_t)(ncol + 16) * C_;
  const float* w2 = inw + (size_t)(ncol + 32) * C_;
  const float* w3 = inw + (size_t)(ncol + 48) * C_;
  v8f a0 = {}, a1 = {}, a2 = {}, a3 = {};
#pragma unroll 4
  for (int k0 = 0; k0 < C_; k0 += 4) {
    int k = k0 + khalf, k1 = k + 1;
    v2f a, b0, b1, b2, b3;
    a.x = urow[k]; a.y = urow[k1];
    b0.x = w0[k]; b0.y = w0[k1];
    b1.x = w1[k]; b1.y = w1[k1];
    b2.x = w2[k]; b2.y = w2[k1];
    b3.x = w3[k]; b3.y = w3[k1];
    a0 = wmma_f32(a, b0, a0);
    a1 = wmma_f32(a, b1, a1);
    a2 = wmma_f32(a, b2, a2);
    a3 = wmma_f32(a, b3, a3);
  }
  int mbase = mt * 16 + ((lane >> 4) ? 8 : 0);
  int nb = ng * 64 + (lane & 15);
#pragma unroll
  for (int r = 0; r < 8; ++r) {
    size_t row = (size_t)(mbase + r) * (2 * D_);
    xz[row + nb +  0] = a0[r];
    xz[row + nb + 16] = a1[r];
    xz[row + nb + 32] = a2[r];
    xz[row + nb + 48] = a3[r];
  }
}

// ---------------------------------------------------------------------------
// K4: depthwise 3x3 conv + SiLU -> xconv (B, D, H, W)
// ---------------------------------------------------------------------------
__global__ __launch_bounds__(256) void k_dwconv(
    const float* __restrict__ xz, const float* __restrict__ cw,
    const float* __restrict__ cb, float* __restrict__ xconv) {
  int idx = blockIdx.x * blockDim.x + threadIdx.x;   // b*D*L + d*L + l
  int b = idx >> 19;
  int rem = idx & ((1 << 19) - 1);
  int d = rem >> 10;
  int l = rem & 1023;
  int h = l >> 5, w = l & 31;
  float s = cb[d];
#pragma unroll
  for (int kh = 0; kh < 3; ++kh) {
    int hh = h + kh - 1;
    if (hh < 0 || hh >= H_) continue;
#pragma unroll
    for (int kw = 0; kw < 3; ++kw) {
      int ww = w + kw - 1;
      if (ww < 0 || ww >= W_) continue;
      s += cw[d * 9 + kh * 3 + kw] *
           xz[((size_t)(b * L_ + (hh << 5) + ww)) * (2 * D_) + d];
    }
  }
  xconv[idx] = s * sigmoidf_(s);
}

// ---------------------------------------------------------------------------
// K5: x_dbl GEMM  per (b,k): [48,1024] = xproj_w[k] (48x512) * xs (512x1024)
// One wave computes the full 48M x 16N column strip (3 accumulators,
// shared B gather -- the expensive operand here).
// ---------------------------------------------------------------------------
__global__ __launch_bounds__(256) void k_xdbl(
    const float* __restrict__ xconv, const float* __restrict__ xproj,
    float* __restrict__ xdbl) {
  int wave = blockIdx.x * (blockDim.x >> 5) + (threadIdx.x >> 5);
  int lane = threadIdx.x & 31;
  int nt = wave & 63;                // 0..63
  int bk = wave >> 6;                // 0..7
  int k  = bk & 3;
  int b  = bk >> 2;
  int khalf = (lane >> 4) << 1;
  int ncol  = nt * 16 + (lane & 15);
  const float* ar0 = xproj + ((size_t)k * 48 +  0 + (lane & 15)) * D_;
  const float* ar1 = xproj + ((size_t)k * 48 + 16 + (lane & 15)) * D_;
  const float* ar2 = xproj + ((size_t)k * 48 + 32 + (lane & 15)) * D_;
  v8f a0 = {}, a1 = {}, a2 = {};
#pragma unroll 4
  for (int k0 = 0; k0 < D_; k0 += 4) {
    int kk = k0 + khalf, kk1 = kk + 1;
    v2f bf, f0, f1, f2;
    bf.x = xs_val(xconv, b, k, kk,  ncol);
    bf.y = xs_val(xconv, b, k, kk1, ncol);
    f0.x = ar0[kk]; f0.y = ar0[kk1];
    f1.x = ar1[kk]; f1.y = ar1[kk1];
    f2.x = ar2[kk]; f2.y = ar2[kk1];
    a0 = wmma_f32(f0, bf, a0);
    a1 = wmma_f32(f1, bf, a1);
    a2 = wmma_f32(f2, bf, a2);
  }
  int mhalf = (lane >> 4) ? 8 : 0;
  int n = nt * 16 + (lane & 15);
  size_t base = (size_t)bk * 48 * L_;
#pragma unroll
  for (int r = 0; r < 8; ++r) {
    xdbl[base + (size_t)( 0 + mhalf + r) * L_ + n] = a0[r];
    xdbl[base + (size_t)(16 + mhalf + r) * L_ + n] = a1[r];
    xdbl[base + (size_t)(32 + mhalf + r) * L_ + n] = a2[r];
  }
}

// ---------------------------------------------------------------------------
// K6: dts GEMM + softplus  per (b,k): [512,1024] = dt_w[k](512x16) * dts_r
// 16M x 64N per wave, K=16 fully unrolled.
// ---------------------------------------------------------------------------
__global__ __launch_bounds__(256) void k_dts(
    const float* __restrict__ xdbl, const float* __restrict__ dtw,
    const float* __restrict__ dtb, float* __restrict__ dts) {
  int wave = blockIdx.x * (blockDim.x >> 5) + (threadIdx.x >> 5);
  int lane = threadIdx.x & 31;
  int ng = wave & 15;  wave >>= 4;   // 0..15
  int mt = wave & 31;  wave >>= 5;   // 0..31
  int k  = wave & 3;
  int b  = wave >> 2;
  int mrowA = mt * 16 + (lane & 15);
  int khalf = (lane >> 4) << 1;
  int ncol  = ng * 64 + (lane & 15);
  const float* arow = dtw + ((size_t)k * D_ + mrowA) * R_;
  const float* bb   = xdbl + (size_t)(b * K_ + k) * 48 * L_;
  v8f a0 = {}, a1 = {}, a2 = {}, a3 = {};
#pragma unroll
  for (int k0 = 0; k0 < R_; k0 += 4) {
    int kk = k0 + khalf, kk1 = kk + 1;
    v2f a, b0, b1, b2, b3;
    a.x = arow[kk]; a.y = arow[kk1];
    const float* r0 = bb + (size_t)kk  * L_ + ncol;
    const float* r1 = bb + (size_t)kk1 * L_ + ncol;
    b0.x = r0[ 0]; b0.y = r1[ 0];
    b1.x = r0[16]; b1.y = r1[16];
    b2.x = r0[32]; b2.y = r1[32];
    b3.x = r0[48]; b3.y = r1[48];
    a0 = wmma_f32(a, b0, a0);
    a1 = wmma_f32(a, b1, a1);
    a2 = wmma_f32(a, b2, a2);
    a3 = wmma_f32(a, b3, a3);
  }
  int mbase = mt * 16 + ((lane >> 4) ? 8 : 0);
  int nb = ng * 64 + (lane & 15);
  size_t obase = (size_t)(b * K_ + k) * D_ * L_;
#pragma unroll
  for (int r = 0; r < 8; ++r) {
    int m = mbase + r;
    float bias = dtb[k * D_ + m];
    size_t row = obase + (size_t)m * L_;
    dts[row + nb +  0] = softplusf_(a0[r] + bias);
    dts[row + nb + 16] = softplusf_(a1[r] + bias);
    dts[row + nb + 32] = softplusf_(a2[r] + bias);
    dts[row + nb + 48] = softplusf_(a3[r] + bias);
  }
}

// ---------------------------------------------------------------------------
// K7: selective scan. Block = (b,k,dquarter): 128 threads, 32 blocks.
// B/C state rows staged in LDS with async global->LDS copies (ASYNCcnt).
// ---------------------------------------------------------------------------
__global__ __launch_bounds__(128) void k_scan(
    const float* __restrict__ xdbl, const float* __restrict__ dts,
    const float* __restrict__ xconv, const float* __restrict__ A_log,
    const float* __restrict__ Ds, float* __restrict__ ys) {
  __shared__ float Bs_s[N_][64];
  __shared__ float Cs_s[N_][64];
  int bk = blockIdx.x >> 2;
  int dq = blockIdx.x & 3;
  int b = bk >> 2, k = bk & 3;
  int d = dq * 128 + threadIdx.x;
  float A[N_], h[N_];
#pragma unroll
  for (int n = 0; n < N_; ++n) {
    A[n] = -__expf(A_log[(k * D_ + d) * N_ + n]);
    h[n] = 0.0f;
  }
  float Dval = Ds[k * D_ + d];
  const float* bc = xdbl + (size_t)bk * 48 * L_;
  const float* dtrow = dts + ((size_t)bk * D_ + d) * L_;
  float* yrow = ys + ((size_t)bk * D_ + d) * L_;
  unsigned ldsB = (unsigned)(size_t)(&Bs_s[0][0]);
  unsigned ldsC = (unsigned)(size_t)(&Cs_s[0][0]);

  for (int l0 = 0; l0 < L_; l0 += 64) {
    // async-copy B rows (16..31) and C rows (32..47) chunks into LDS
    for (int i = threadIdx.x; i < N_ * 64; i += 128) {
      int n = i >> 6, j = i & 63;
      unsigned dB = ldsB + ((unsigned)i << 2);
      unsigned dC = ldsC + ((unsigned)i << 2);
      const float* sB = bc + (size_t)(R_ + n) * L_ + l0 + j;
      const float* sC = bc + (size_t)(R_ + N_ + n) * L_ + l0 + j;
      asm volatile("global_load_async_to_lds_b32 %0, %1, off"
                   :: "v"(dB), "v"(sB) : "memory");
      asm volatile("global_load_async_to_lds_b32 %0, %1, off"
                   :: "v"(dC), "v"(sC) : "memory");
    }
    asm volatile("s_wait_asynccnt 0x0" ::: "memory");
    __syncthreads();
    for (int j = 0; j < 64; ++j) {
      int l = l0 + j;
      float dt = dtrow[l];
      float xv = xs_val(xconv, b, k, d, l);
      float dx = dt * xv;
      float y = Dval * xv;
#pragma unroll
      for (int n = 0; n < N_; ++n) {
        float dA = __expf(dt * A[n]);
        h[n] = dA * h[n] + dx * Bs_s[n][j];
        y += h[n] * Cs_s[n][j];
      }
      yrow[l] = y;
    }
    __syncthreads();
  }
}

// ---------------------------------------------------------------------------
// K8: combine 4 directions + LN(512) + SiLU(z) gate -> yg (2048,512)
// ---------------------------------------------------------------------------
__global__ __launch_bounds__(512) void k_combine(
    const float* __restrict__ ys, const float* __restrict__ xz,
    const float* __restrict__ g, const float* __restrict__ bb,
    float* __restrict__ yg) {
  __shared__ float red[512];
  int b = blockIdx.x >> 10;
  int l = blockIdx.x & 1023;
  int d = threadIdx.x;
  int h = l >> 5, w = l & 31;
  int lT = (w << 5) | h;
  size_t base = (size_t)(b * K_) * D_ * L_;
  float v = ys[base + ((size_t)(0 * D_ + d)) * L_ + l]
          + ys[base + ((size_t)(2 * D_ + d)) * L_ + (L_ - 1 - l)]
          + ys[base + ((size_t)(1 * D_ + d)) * L_ + lT]
          + ys[base + ((size_t)(3 * D_ + d)) * L_ + (L_ - 1 - lT)];
  red[d] = v; __syncthreads();
  for (int s = 256; s > 0; s >>= 1) { if (d < s) red[d] += red[d + s]; __syncthreads(); }
  float mu = red[0] * (1.0f / D_);
  __syncthreads();
  float dv = v - mu;
  red[d] = dv * dv; __syncthreads();
  for (int s = 256; s > 0; s >>= 1) { if (d < s) red[d] += red[d + s]; __syncthreads(); }
  float var = red[0] * (1.0f / D_);
  float nrm = dv * rsqrtf(var + 1e-5f) * g[d] + bb[d];
  float zv = xz[(size_t)(b * L_ + l) * (2 * D_) + D_ + d];
  yg[(size_t)(b * L_ + l) * D_ + d] = nrm * zv * sigmoidf_(zv);
}

// ---------------------------------------------------------------------------
// K9: out GEMM + residual  t2[2048,256] = t + yg[2048,512] * out_w^T
// 16M x 64N per wave.
// ---------------------------------------------------------------------------
__global__ __launch_bounds__(256) void k_out(
    const float* __restrict__ yg, const float* __restrict__ ow,
    const float* __restrict__ t, float* __restrict__ t2) {
  int wave = blockIdx.x * (blockDim.x >> 5) + (threadIdx.x >> 5);
  int lane = threadIdx.x & 31;
  int mt = wave >> 2;               // 0..127
  int ng = wave & 3;                // 0..3
  int mrowA = mt * 16 + (lane & 15);
  int khalf = (lane >> 4) << 1;
  int ncol  = ng * 64 + (lane & 15);
  const float* arow = yg + (size_t)mrowA * D_;
  const float* w0 = ow + (size_t)(ncol +  0) * D_;
  const float* w1 = ow + (size_t)(ncol + 16) * D_;
  const float* w2 = ow + (size_t)(ncol + 32) * D_;
  const float* w3 = ow + (size_t)(ncol + 48) * D_;
  v8f a0 = {}, a1 = {}, a2 = {}, a3 = {};
#pragma unroll 4
  for (int k0 = 0; k0 < D_; k0 += 4) {
    int k = k0 + khalf, k1 = k + 1;
    v2f a, b0, b1, b2, b3;
    a.x = arow[k]; a.y = arow[k1];
    b0.x = w0[k]; b0.y = w0[k1];
    b1.x = w1[k]; b1.y = w1[k1];
    b2.x = w2[k]; b2.y = w2[k1];
    b3.x = w3[k]; b3.y = w3[k1];
    a0 = wmma_f32(a, b0, a0);
    a1 = wmma_f32(a, b1, a1);
    a2 = wmma_f32(a, b2, a2);
    a3 = wmma_f32(a, b3, a3);
  }
  int mbase = mt * 16 + ((lane >> 4) ? 8 : 0);
  int nb = ng * 64 + (lane & 15);
#pragma unroll
  for (int r = 0; r < 8; ++r) {
    size_t row = (size_t)(mbase + r) * C_;
    t2[row + nb +  0] = t[row + nb +  0] + a0[r];
    t2[row + nb + 16] = t[row + nb + 16] + a1[r];
    t2[row + nb + 32] = t[row + nb + 32] + a2[r];
    t2[row + nb + 48] = t[row + nb + 48] + a3[r];
  }
}

// ---------------------------------------------------------------------------
// K10: mean-pool over (H,W) + linear head -> out (B,)
// ---------------------------------------------------------------------------
__global__ __launch_bounds__(256) void k_head(
    const float* __restrict__ t2, const float* __restrict__ hw,
    const float* __restrict__ hb, float* __restrict__ out) {
  __shared__ float red[256];
  int b = blockIdx.x, c = threadIdx.x;
  float s = 0.0f;
  for (int l = 0; l < L_; ++l)
    s += t2[(size_t)(b * L_ + l) * C_ + c];
  red[c] = s * hw[c];
  __syncthreads();
  for (int sh = 128; sh > 0; sh >>= 1) { if (c < sh) red[c] += red[c + sh]; __syncthreads(); }
  if (c == 0) out[b] = red[0] * (1.0f / L_) + hb[0];
}

// ---------------------------------------------------------------------------
extern "C" void kernel_launch(void* const* d_in, const int* in_sizes, int n_in,
                              void* d_out, int out_size, void* d_ws, size_t ws_size,
                              hipStream_t stream) {
  const float* x       = (const float*)d_in[0];
  const float* patch_w = (const float*)d_in[1];
  const float* patch_b = (const float*)d_in[2];
  const float* ln_g    = (const float*)d_in[3];
  const float* ln_b    = (const float*)d_in[4];
  const float* in_w    = (const float*)d_in[5];
  const float* conv_w  = (const float*)d_in[6];
  const float* conv_b  = (const float*)d_in[7];
  const float* xproj_w = (const float*)d_in[8];
  const float* dt_w    = (const float*)d_in[9];
  const float* dt_b    = (const float*)d_in[10];
  const float* A_log   = (const float*)d_in[11];
  const float* Ds      = (const float*)d_in[12];
  const float* on_g    = (const float*)d_in[13];
  const float* on_b    = (const float*)d_in[14];
  const float* out_w   = (const float*)d_in[15];
  const float* head_w  = (const float*)d_in[16];
  const float* head_b  = (const float*)d_in[17];

  float* ws    = (float*)d_ws;
  float* t     = ws;                 // 2048*256
  float* u     = t     + 524288;     // 2048*256
  float* xz    = u     + 524288;     // 2048*1024
  float* xconv = xz    + 2097152;    // 2*512*1024
  float* xdbl  = xconv + 1048576;    // 2*4*48*1024
  float* dts   = xdbl  + 393216;     // 2*4*512*1024
  float* ys    = dts   + 4194304;    // 2*4*512*1024
  float* yg    = ys    + 4194304;    // 2048*512
  float* t2    = yg    + 1048576;    // 2048*256

  k_patch_embed<<<64, 256, 0, stream>>>(x, patch_w, patch_b, t);
  k_ln256     <<<2048, 256, 0, stream>>>(t, ln_g, ln_b, u);
  k_xz        <<<256, 256, 0, stream>>>(u, in_w, xz);
  k_dwconv    <<<4096, 256, 0, stream>>>(xz, conv_w, conv_b, xconv);
  k_xdbl      <<<64, 256, 0, stream>>>(xconv, xproj_w, xdbl);
  k_dts       <<<512, 256, 0, stream>>>(xdbl, dt_w, dt_b, dts);
  k_scan      <<<32, 128, 0, stream>>>(xdbl, dts, xconv, A_log, Ds, ys);
  k_combine   <<<2048, 512, 0, stream>>>(ys, xz, on_g, on_b, yg);
  k_out       <<<64, 256, 0, stream>>>(yg, out_w, t, t2);
  k_head      <<<2, 256, 0, stream>>>(t2, head_w, head_b, (float*)d_out);
}